// FPNPostProcessor_19086834663979
// MI455X (gfx1250) — compile-verified
//
#include <hip/hip_runtime.h>
#include <math.h>

// ---------------------------------------------------------------------------
// FPN post-processor for MI455X (gfx1250, wave32).
//   d_in[0] class_logits   [8192][81]  f32
//   d_in[1] box_regression [8192][324] f32
//   d_in[2] proposals      [8192][4]   f32
//   d_in[3] height (int scalar), d_in[4] width (int scalar)
//   d_out = scores[8][100] ++ boxes[8][100][4] ++ labels[8][100]  (4800 f32)
// Workspace layout:
//   scores_t : [8][80][1024]   f32  (2.62 MB)   class-major per image
//   boxes_t  : [8][80][1024]   f32x4 (10.5 MB)
// gfx1250 paths exercised: GLOBAL_LOAD_ASYNC_TO_LDS_B128 (ASYNCcnt),
// TENSOR_LOAD_TO_LDS via TDM descriptor (TENSORcnt), wave32 shfl reductions.
// No WMMA by design: the workload has zero matmul content.
// ---------------------------------------------------------------------------

#define NUM_CLASSES 81
#define FG          80
#define NPI         1024
#define NIMG        8
#define MROWS       (NIMG * NPI)
#define SCORE_THRESH 0.05f
#define NMS_THRESH   0.5f
#define DETS         100

typedef unsigned int u32x4 __attribute__((ext_vector_type(4)));
typedef int i32x4 __attribute__((ext_vector_type(4)));
typedef int i32x8 __attribute__((ext_vector_type(8)));

// -------------------- CDNA5 async global->LDS helpers ----------------------
__device__ __forceinline__ void async_ld_b128(unsigned lds_off, unsigned goff,
                                              unsigned long long gbase) {
  asm volatile("global_load_async_to_lds_b128 %0, %1, %2"
               :
               : "v"(lds_off), "v"(goff), "s"(gbase)
               : "memory");
}
__device__ __forceinline__ void wait_asynccnt0() {
  asm volatile("s_wait_asynccnt 0" ::: "memory");
}

// -------------------- CDNA5 Tensor Data Mover (TDM) ------------------------
// Copies a contiguous tile of `words` DWORDs from gaddr into LDS byte offset
// lds_off as a 2D tensor (dim0 = words, dim1 = 1).  One issue per wave;
// tracked by TENSORcnt.
#if __has_builtin(__builtin_amdgcn_tensor_load_to_lds)
#define HAVE_TDM 1
__device__ __forceinline__ void tdm_load_tile(unsigned lds_off,
                                              unsigned long long gaddr,
                                              unsigned words) {
  // D# group 0: count=1 | lds_addr | global_addr(57b) | type=2
  u32x4 g0;
  g0.x = 1u;                                    // count=1, user descriptor
  g0.y = lds_off;                               // lds_addr (bytes)
  g0.z = (unsigned)gaddr;                       // global_addr[31:0]
  g0.w = ((unsigned)(gaddr >> 32) & 0x01FFFFFFu) | 0x80000000u;  // [56:32],type=2
  // D# group 1: data_size=4B, tensor_dim0=tile_dim0=words, dim1=1, stride=words
  i32x8 g1;
  g1[0] = 0x00020000;            // workgroup_mask=0, data_size=2 (4 bytes)
  g1[1] = (int)(words << 16);    // tensor_dim0[15:0] in [31:16]
  g1[2] = 0x00010000 | (int)(words >> 16);  // tensor_dim0[31:16], tensor_dim1=1
  g1[3] = (int)((words & 0xFFFFu) << 16);   // tile_dim0 (16b)
  g1[4] = 1;                     // tile_dim1=1, tile_dim2=0
  g1[5] = (int)words;            // tensor_dim0_stride[31:0]
  g1[6] = 0;                     // stride hi, tensor_dim1_stride lo
  g1[7] = 0;
  i32x4 gz = {0, 0, 0, 0};       // groups 2/3: dims >=3 unused
#if __has_include(<hip/amd_detail/amd_gfx1250_TDM.h>)
  i32x8 gz8 = {0, 0, 0, 0, 0, 0, 0, 0};  // therock toolchain: 6-arg builtin
  __builtin_amdgcn_tensor_load_to_lds(g0, g1, gz, gz, gz8, 0);
#else
  __builtin_amdgcn_tensor_load_to_lds(g0, g1, gz, gz, 0);  // ROCm 7.2: 5-arg
#endif
}
#else
#define HAVE_TDM 0
#endif

// ---------------------------------------------------------------------------
// Kernel 1: fused softmax + box decode + clip, transposed store.
// One wave32 per proposal row; 8 waves (256 threads) per block.
// ---------------------------------------------------------------------------
__global__ __launch_bounds__(256) void softmax_decode_kernel(
    const float* __restrict__ logits, const float* __restrict__ regr,
    const float* __restrict__ props, const int* __restrict__ hptr,
    const int* __restrict__ wptr, float* __restrict__ scores_t,
    float4* __restrict__ boxes_t) {
  const int wave = threadIdx.x >> 5;
  const int lane = threadIdx.x & 31;
  const int row = blockIdx.x * 8 + wave;
  if (row >= MROWS) return;

  const float* lrow = logits + (size_t)row * NUM_CLASSES;
  float v0 = lrow[lane];
  float v1 = lrow[lane + 32];
  float v2 = (lane + 64 < NUM_CLASSES) ? lrow[lane + 64] : -1e30f;

  float m = fmaxf(v0, fmaxf(v1, v2));
  for (int off = 16; off > 0; off >>= 1) m = fmaxf(m, __shfl_xor(m, off, 32));
  float e0 = __expf(v0 - m);
  float e1 = __expf(v1 - m);
  float e2 = (lane + 64 < NUM_CLASSES) ? __expf(v2 - m) : 0.0f;
  float s = e0 + e1 + e2;
  for (int off = 16; off > 0; off >>= 1) s += __shfl_xor(s, off, 32);
  const float inv = 1.0f / s;

  const float4 p = ((const float4*)props)[row];
  const float w = p.z - p.x + 1.0f;
  const float h = p.w - p.y + 1.0f;
  const float cx = p.x + 0.5f * w;
  const float cy = p.y + 0.5f * h;
  const float W1 = (float)(wptr[0] - 1);
  const float H1 = (float)(hptr[0] - 1);
  const float CLIP = 4.135166556742356f;  // log(1000/16)

  const int img = row >> 10;
  const int n = row & (NPI - 1);
  const float* rrow = regr + (size_t)row * 4 * NUM_CLASSES;
  float ev[3] = {e0, e1, e2};

#pragma unroll
  for (int k = 0; k < 3; ++k) {
    const int c = lane + k * 32;
    if (c < 1 || c >= NUM_CLASSES) continue;  // drop background class 0
    const float prob = ev[k] * inv;
    const float dx = rrow[c * 4 + 0] * 0.1f;
    const float dy = rrow[c * 4 + 1] * 0.1f;
    const float dw = fminf(rrow[c * 4 + 2] * 0.2f, CLIP);
    const float dh = fminf(rrow[c * 4 + 3] * 0.2f, CLIP);
    const float px = dx * w + cx;
    const float py = dy * h + cy;
    const float pw = __expf(dw) * w;
    const float ph = __expf(dh) * h;
    float x1 = px - 0.5f * pw;
    float y1 = py - 0.5f * ph;
    float x2 = px + 0.5f * pw - 1.0f;
    float y2 = py + 0.5f * ph - 1.0f;
    x1 = fminf(fmaxf(x1, 0.0f), W1);
    x2 = fminf(fmaxf(x2, 0.0f), W1);
    y1 = fminf(fmaxf(y1, 0.0f), H1);
    y2 = fminf(fmaxf(y2, 0.0f), H1);
    const int idx = (img * FG + (c - 1)) * NPI + n;
    scores_t[idx] = (prob > SCORE_THRESH) ? prob : 0.0f;  // pre-threshold
    boxes_t[idx] = make_float4(x1, y1, x2, y2);
  }
}

// ---------------------------------------------------------------------------
// Kernel 2: per-(image,class) sort + greedy NMS, fully LDS-resident.
// 640 blocks x 256 threads.  48 KB static LDS.
// Boxes tile (16 KB) arrives via TDM; scores tile (4 KB) via async-to-LDS.
// ---------------------------------------------------------------------------
#define OFF_KEY 0u       // 1024 f32 sort keys (scores)        4 KB
#define OFF_BOX 4096u    // 1024 float4 raw boxes             16 KB
#define OFF_IDX 20480u   // 1024 i32 payload indices           4 KB
#define OFF_SBX 24576u   // 1024 float4 sorted boxes          16 KB
#define OFF_ARE 40960u   // 1024 f32 areas of sorted boxes     4 KB
#define OFF_KEP 45056u   // 1024 i32 keep flags                4 KB
#define SMEM_BYTES 49152u

__global__ __launch_bounds__(256) void nms_kernel(float* __restrict__ scores_t,
                                                  float4* __restrict__ boxes_t) {
  __shared__ __align__(16) unsigned char smem[SMEM_BYTES];
  float* key = (float*)(smem + OFF_KEY);
  float4* rbox = (float4*)(smem + OFF_BOX);
  int* pidx = (int*)(smem + OFF_IDX);
  float4* sbox = (float4*)(smem + OFF_SBX);
  float* sarea = (float*)(smem + OFF_ARE);
  int* keep = (int*)(smem + OFF_KEP);
  __shared__ int Vsh;

  const int tid = threadIdx.x;
  const int rowbase = blockIdx.x * NPI;

  // --- tile staging: scores on ASYNCcnt, boxes on TENSORcnt (TDM DMA) ---
  {
    const unsigned long long sbase =
        (unsigned long long)(scores_t + (size_t)rowbase);
    async_ld_b128(OFF_KEY + tid * 16u, tid * 16u, sbase);  // 4 KB scores
    const unsigned long long bbase =
        (unsigned long long)(boxes_t + (size_t)rowbase);
#if HAVE_TDM
    if (threadIdx.x < 32) {  // one wave issues the 16 KB DMA
      tdm_load_tile(OFF_BOX, bbase, NPI * 4u /* dwords */);
      __builtin_amdgcn_s_wait_tensorcnt(0);
    }
#else
#pragma unroll
    for (int k = 0; k < 4; ++k) {  // fallback: async path for boxes
      const unsigned e = (unsigned)(tid + k * 256) * 16u;
      async_ld_b128(OFF_BOX + e, e, bbase);
    }
#endif
    wait_asynccnt0();
  }
  if (tid == 0) Vsh = NPI;
  __syncthreads();

  for (int t = tid; t < NPI; t += 256) pidx[t] = t;

  // --- bitonic sort, descending score, ascending index on ties ---
  for (unsigned k = 2; k <= NPI; k <<= 1) {
    for (unsigned j = k >> 1; j > 0; j >>= 1) {
      __syncthreads();
      for (unsigned t = tid; t < NPI; t += 256) {
        const unsigned x = t ^ j;
        if (x > t) {
          const float ka = key[t], kb = key[x];
          const int ia = pidx[t], ib = pidx[x];
          const bool inorder = (ka > kb) || (ka == kb && ia < ib);
          const bool up = ((t & k) == 0);
          if (inorder != up) {
            key[t] = kb; key[x] = ka;
            pidx[t] = ib; pidx[x] = ia;
          }
        }
      }
    }
  }
  __syncthreads();

  // --- gather sorted boxes, areas, keep flags; find valid count V ---
  for (int t = tid; t < NPI; t += 256) {
    const float4 b = rbox[pidx[t]];
    sbox[t] = b;
    sarea[t] = (b.z - b.x + 1.0f) * (b.w - b.y + 1.0f);
    const float kt = key[t];
    keep[t] = (kt > SCORE_THRESH) ? 1 : 0;
    if (kt <= SCORE_THRESH && (t == 0 || key[t - 1] > SCORE_THRESH)) Vsh = t;
  }
  __syncthreads();
  const int V = Vsh;  // boundary of scores > thresh (sorted), uniform

  // --- greedy suppression (exact match of the reference fori_loop) ---
  for (int i = 0; i < V; ++i) {
    __syncthreads();
    if (keep[i]) {  // uniform branch: same LDS word for all lanes
      const float4 b = sbox[i];
      const float ai = sarea[i];
      for (int j = tid; j < V; j += 256) {  // entries >= V already invalid
        if (j > i && keep[j]) {
          const float4 c = sbox[j];
          const float xx1 = fmaxf(b.x, c.x);
          const float yy1 = fmaxf(b.y, c.y);
          const float xx2 = fminf(b.z, c.z);
          const float yy2 = fminf(b.w, c.w);
          const float iw = fmaxf(xx2 - xx1 + 1.0f, 0.0f);
          const float ih = fmaxf(yy2 - yy1 + 1.0f, 0.0f);
          const float inter = iw * ih;
          const float iou = inter / (ai + sarea[j] - inter);
          if (iou > NMS_THRESH) keep[j] = 0;
        }
      }
    }
  }
  __syncthreads();

  // --- write back: suppressed scores zeroed, boxes kept in sorted order ---
  for (int t = tid; t < NPI; t += 256) {
    scores_t[rowbase + t] = keep[t] ? key[t] : 0.0f;
    boxes_t[rowbase + t] = sbox[t];
  }
}

// ---------------------------------------------------------------------------
// Kernel 3: per-image top-100 (iterative argmax, lowest-index tie-break to
// match lax.top_k).  8 blocks x 256 threads; 81920 candidates per image
// (fully L2-resident at 192 MB).
// ---------------------------------------------------------------------------
__global__ __launch_bounds__(256) void topk_kernel(
    float* __restrict__ scores_t, const float4* __restrict__ boxes_t,
    float* __restrict__ out) {
  __shared__ float bv[256];
  __shared__ int bi[256];
  const int tid = threadIdx.x;
  const int img = blockIdx.x;
  float* sc = scores_t + (size_t)img * FG * NPI;
  const float4* bx = boxes_t + (size_t)img * FG * NPI;

  float* out_scores = out;                    // [8][100]
  float* out_boxes = out + NIMG * DETS;       // [8][100][4]
  float* out_labels = out + NIMG * DETS * 5;  // [8][100]

  for (int k = 0; k < DETS; ++k) {
    float best = -2.0f;
    int bidx = 0x7fffffff;
    for (int i = tid; i < FG * NPI; i += 256) {
      const float v = sc[i];
      if (v > best || (v == best && i < bidx)) { best = v; bidx = i; }
    }
    bv[tid] = best;
    bi[tid] = bidx;
    __syncthreads();
    for (int s = 128; s > 0; s >>= 1) {
      if (tid < s) {
        const float v2 = bv[tid + s];
        const int i2 = bi[tid + s];
        if (v2 > bv[tid] || (v2 == bv[tid] && i2 < bi[tid])) {
          bv[tid] = v2; bi[tid] = i2;
        }
      }
      __syncthreads();
    }
    if (tid == 0) {
      const float v = bv[0];
      const int id = bi[0];
      const bool ok = v > SCORE_THRESH;
      out_scores[img * DETS + k] = ok ? v : 0.0f;
      const float4 b = bx[id];
      float* ob = out_boxes + (size_t)(img * DETS + k) * 4;
      ob[0] = ok ? b.x : 0.0f;
      ob[1] = ok ? b.y : 0.0f;
      ob[2] = ok ? b.z : 0.0f;
      ob[3] = ok ? b.w : 0.0f;
      out_labels[img * DETS + k] = ok ? (float)(id / NPI + 1) : 0.0f;
      sc[id] = -1.0f;  // remove from further rounds
    }
    __syncthreads();
  }
}

// ---------------------------------------------------------------------------
extern "C" void kernel_launch(void* const* d_in, const int* in_sizes, int n_in,
                              void* d_out, int out_size, void* d_ws,
                              size_t ws_size, hipStream_t stream) {
  (void)in_sizes; (void)n_in; (void)out_size; (void)ws_size;
  const float* logits = (const float*)d_in[0];
  const float* regr = (const float*)d_in[1];
  const float* props = (const float*)d_in[2];
  const int* hptr = (const int*)d_in[3];
  const int* wptr = (const int*)d_in[4];

  float* scores_t = (float*)d_ws;  // 2.62 MB
  float4* boxes_t =
      (float4*)((char*)d_ws + (size_t)NIMG * FG * NPI * sizeof(float));

  softmax_decode_kernel<<<MROWS / 8, 256, 0, stream>>>(
      logits, regr, props, hptr, wptr, scores_t, boxes_t);
  nms_kernel<<<NIMG * FG, 256, 0, stream>>>(scores_t, boxes_t);
  topk_kernel<<<NIMG, 256, 0, stream>>>(scores_t, boxes_t, (float*)d_out);
}